// TransitionDown_56624848831036
// MI455X (gfx1250) — compile-verified
//
#include <hip/hip_runtime.h>

#define BATCH   4
#define NPTS    8192
#define NPOINT  2048
#define NSAMPLE 16
#define INPL    64
#define OUTPL   128
#define KF      67          // 64 px features + 3 xyz-diff
#define KPAD    68          // padded to multiple of 4 for 16x16x4 WMMA
#define MTOT    (BATCH * NPOINT * NSAMPLE)   // 131072 GEMM rows
#define OUT_XYZ_OFF (BATCH * NPOINT * OUTPL) // new_px elements before new_xyz

typedef float v2f __attribute__((ext_vector_type(2)));
typedef float v8f __attribute__((ext_vector_type(8)));

// ---------------------------------------------------------------------------
// 1) Farthest point sampling: one 1024-thread workgroup per batch.
//    Distance array lives in LDS; sequential 2048-round argmax with
//    lowest-index tie-break (matches jnp.argmax). Emits new_xyz directly.
// ---------------------------------------------------------------------------
__global__ __launch_bounds__(1024) void fps_kernel(const float* __restrict__ xyz,
                                                   float* __restrict__ new_xyz) {
  const int b = blockIdx.x;
  const int tid = threadIdx.x;
  const float* x = xyz + (size_t)b * NPTS * 3;
  __shared__ float dist[NPTS];   // 32 KB
  __shared__ float rv[1024];     //  4 KB
  __shared__ int   ri[1024];     //  4 KB
  __shared__ int   curf;
  for (int i = tid; i < NPTS; i += 1024) dist[i] = 1e10f;
  if (tid == 0) curf = 0;
  __syncthreads();
  for (int it = 0; it < NPOINT; ++it) {
    const int f = curf;
    const float cx = x[f * 3 + 0], cy = x[f * 3 + 1], cz = x[f * 3 + 2];
    if (tid == 0) {  // scan emits the *current* farthest each step
      float* o = new_xyz + ((size_t)b * NPOINT + it) * 3;
      o[0] = cx; o[1] = cy; o[2] = cz;
    }
    float bv = -1.0f; int bi = NPTS;
    for (int i = tid; i < NPTS; i += 1024) {
      float dx = x[i * 3 + 0] - cx, dy = x[i * 3 + 1] - cy, dz = x[i * 3 + 2] - cz;
      float d  = dx * dx + dy * dy + dz * dz;
      float nd = fminf(dist[i], d);
      dist[i] = nd;
      if (nd > bv) { bv = nd; bi = i; }  // ascending i => first max kept
    }
    rv[tid] = bv; ri[tid] = bi;
    __syncthreads();
    for (int s = 512; s > 0; s >>= 1) {
      if (tid < s) {
        float ov = rv[tid + s]; int oi = ri[tid + s];
        if (ov > rv[tid] || (ov == rv[tid] && oi < ri[tid])) { rv[tid] = ov; ri[tid] = oi; }
      }
      __syncthreads();
    }
    if (tid == 0) curf = ri[0];
    __syncthreads();
  }
}

// ---------------------------------------------------------------------------
// 2) KNN: one thread per query centroid; xyz streamed through LDS in 2048-pt
//    chunks; fully unrolled 16-deep register insertion (low-index tie win).
// ---------------------------------------------------------------------------
#define KNN_CH 2048
__global__ __launch_bounds__(256) void knn_kernel(const float* __restrict__ xyz,
                                                  const float* __restrict__ new_xyz,
                                                  int* __restrict__ knn_idx) {
  const int b    = blockIdx.x >> 3;   // 8 query tiles of 256 per batch
  const int tile = blockIdx.x & 7;
  const int tid  = threadIdx.x;
  const int m    = tile * 256 + tid;
  const float* x = xyz + (size_t)b * NPTS * 3;
  const float* q = new_xyz + ((size_t)b * NPOINT + m) * 3;
  const float qx = q[0], qy = q[1], qz = q[2];
  __shared__ float sx[KNN_CH], sy[KNN_CH], sz[KNN_CH];   // 24 KB
  float bd[NSAMPLE]; int bi[NSAMPLE];
#pragma unroll
  for (int j = 0; j < NSAMPLE; ++j) { bd[j] = 3.0e38f; bi[j] = 0; }
  for (int c0 = 0; c0 < NPTS; c0 += KNN_CH) {
    __syncthreads();
    for (int i = tid; i < KNN_CH; i += 256) {
      sx[i] = x[(c0 + i) * 3 + 0];
      sy[i] = x[(c0 + i) * 3 + 1];
      sz[i] = x[(c0 + i) * 3 + 2];
    }
    __syncthreads();
    for (int i = 0; i < KNN_CH; ++i) {
      float dx = sx[i] - qx, dy = sy[i] - qy, dz = sz[i] - qz;
      float d = dx * dx + dy * dy + dz * dz;
      if (d < bd[NSAMPLE - 1]) {
        int gi = c0 + i;
#pragma unroll
        for (int j = NSAMPLE - 1; j >= 0; --j) {
          if (d < bd[j]) {
            if (j < NSAMPLE - 1) { bd[j + 1] = bd[j]; bi[j + 1] = bi[j]; }
            bd[j] = d; bi[j] = gi;
          }
        }
      }
    }
  }
  int* o = knn_idx + ((size_t)b * NPOINT + m) * NSAMPLE;
#pragma unroll
  for (int j = 0; j < NSAMPLE; ++j) o[j] = bi[j];
}

// ---------------------------------------------------------------------------
// 3) Fused gather + GEMM via V_WMMA_F32_16X16X4_F32 (exact fp32, native
//    CDNA5 shape). Block = 256 threads = 8 waves; block tile = 128 rows x 128
//    cols. A tile (gathered [px | xyz-diff] features) staged in LDS.
//    Fragment layouts per ISA 7.12.2:
//      A 16x4:  lanes 0-15 -> K={k0,k0+1}, lanes 16-31 -> K={k0+2,k0+3}
//      B 4x16:  mirrored over N (col = lane&15)
//      C/D:     VGPR r -> M=r (lanes<16) / M=r+8 (lanes>=16), N = lane&15
// ---------------------------------------------------------------------------
__global__ __launch_bounds__(256) void gemm_wmma_kernel(
    const float* __restrict__ px, const float* __restrict__ xyz,
    const float* __restrict__ new_xyz, const int* __restrict__ knn_idx,
    const float* __restrict__ W, const float* __restrict__ bias,
    float* __restrict__ h) {
  __shared__ float As[128][KPAD];   // 34.8 KB
  const int tid = threadIdx.x;
  const int rbase = blockIdx.x * 128;
  // Stage the gathered A tile.
  for (int e = tid; e < 128 * KPAD; e += 256) {
    int row = e / KPAD, k = e - row * KPAD;
    int gr  = rbase + row;
    int b   = gr >> 15;            // / (NPOINT*NSAMPLE) == 32768
    int rem = gr & 32767;
    int m = rem >> 4, s = rem & 15;
    int pi = knn_idx[((size_t)b * NPOINT + m) * NSAMPLE + s];
    float v;
    if (k < INPL) {
      v = px[((size_t)b * NPTS + pi) * INPL + k];
    } else if (k < KF) {
      int c = k - INPL;
      v = xyz[((size_t)b * NPTS + pi) * 3 + c]
        - new_xyz[((size_t)b * NPOINT + m) * 3 + c];
    } else {
      v = 0.0f;                    // K padding
    }
    As[row][k] = v;
  }
  __syncthreads();

  const int w    = tid >> 5;
  const int lane = tid & 31;
  const int hl   = lane & 15;
  const int koff = (lane >> 4) * 2;
  const int rowA = w * 16 + hl;

  v8f acc[8] = {};
  for (int k0 = 0; k0 < KPAD; k0 += 4) {
    v2f a;
    a.x = As[rowA][k0 + koff];
    a.y = As[rowA][k0 + koff + 1];
#pragma unroll
    for (int nt = 0; nt < 8; ++nt) {
      int col = nt * 16 + hl;
      int kb  = k0 + koff;
      v2f bf;
      bf.x = (kb     < KF) ? W[(size_t)kb * OUTPL + col] : 0.0f;
      bf.y = (kb + 1 < KF) ? W[(size_t)(kb + 1) * OUTPL + col] : 0.0f;
      acc[nt] = __builtin_amdgcn_wmma_f32_16x16x4_f32(
          /*neg_a=*/false, a, /*neg_b=*/false, bf,
          /*c_mod=*/(short)0, acc[nt], /*reuse_a=*/false, /*reuse_b=*/false);
    }
  }
  // Epilogue: + bias, write h.
#pragma unroll
  for (int nt = 0; nt < 8; ++nt) {
    int col = nt * 16 + hl;
    float bb = bias[col];
#pragma unroll
    for (int r = 0; r < 8; ++r) {
      int rowg = rbase + w * 16 + ((lane < 16) ? r : r + 8);
      h[(size_t)rowg * OUTPL + col] = acc[nt][r] + bb;
    }
  }
}

// ---------------------------------------------------------------------------
// 4) BatchNorm stats — deterministic two-stage reduction (no float atomics).
// ---------------------------------------------------------------------------
__global__ __launch_bounds__(128) void bn_stage1(const float* __restrict__ h,
                                                 float* __restrict__ psum,
                                                 float* __restrict__ psq) {
  const int c  = threadIdx.x;
  const int r0 = blockIdx.x * 256;   // 512 blocks x 256 rows = 131072
  float s = 0.0f, q = 0.0f;
  for (int r = 0; r < 256; ++r) {
    float v = h[(size_t)(r0 + r) * OUTPL + c];
    s += v; q += v * v;
  }
  psum[blockIdx.x * OUTPL + c] = s;
  psq [blockIdx.x * OUTPL + c] = q;
}

__global__ __launch_bounds__(128) void bn_finalize(const float* __restrict__ psum,
                                                   const float* __restrict__ psq,
                                                   const float* __restrict__ gamma,
                                                   const float* __restrict__ beta,
                                                   float* __restrict__ ab) {
  const int c = threadIdx.x;
  float s = 0.0f, q = 0.0f;
  for (int i = 0; i < 512; ++i) { s += psum[i * OUTPL + c]; q += psq[i * OUTPL + c]; }
  const float inv_m = 1.0f / (float)MTOT;
  float mu  = s * inv_m;
  float var = q * inv_m - mu * mu;     // biased variance (jnp.var default)
  float inv = rsqrtf(var + 1e-5f);
  float A = gamma[c] * inv;
  ab[c]         = A;                   // scale
  ab[OUTPL + c] = beta[c] - mu * A;    // shift
}

// ---------------------------------------------------------------------------
// 5) Normalize + ReLU + max over nsample=16 -> new_px.
// ---------------------------------------------------------------------------
__global__ __launch_bounds__(256) void maxpool_kernel(const float* __restrict__ h,
                                                      const float* __restrict__ ab,
                                                      float* __restrict__ out) {
  int g  = blockIdx.x * 256 + threadIdx.x;  // < BATCH*NPOINT*OUTPL
  int c  = g & (OUTPL - 1);
  int pm = g >> 7;                          // (b*NPOINT + m)
  float A = ab[c], Bc = ab[OUTPL + c];
  float mx = 0.0f;                          // ReLU outputs are >= 0
  for (int s = 0; s < NSAMPLE; ++s) {
    float v = h[((size_t)pm * NSAMPLE + s) * OUTPL + c] * A + Bc;
    mx = fmaxf(mx, fmaxf(v, 0.0f));
  }
  out[g] = mx;
}

// ---------------------------------------------------------------------------
extern "C" void kernel_launch(void* const* d_in, const int* in_sizes, int n_in,
                              void* d_out, int out_size, void* d_ws, size_t ws_size,
                              hipStream_t stream) {
  (void)in_sizes; (void)n_in; (void)out_size; (void)ws_size;
  const float* px    = (const float*)d_in[0];
  const float* xyz   = (const float*)d_in[1];
  const float* W     = (const float*)d_in[2];
  const float* bias  = (const float*)d_in[3];
  const float* gamma = (const float*)d_in[4];
  const float* beta  = (const float*)d_in[5];
  float* out     = (float*)d_out;
  float* new_xyz = out + OUT_XYZ_OFF;

  // Workspace layout (bytes): knn idx 512K | h 64M | psum 256K | psq 256K | ab 1K
  char*  ws   = (char*)d_ws;
  int*   knn  = (int*)ws;
  float* h    = (float*)(ws + 524288);
  float* psum = (float*)(ws + 524288 + 67108864);
  float* psq  = (float*)(ws + 524288 + 67108864 + 262144);
  float* ab   = (float*)(ws + 524288 + 67108864 + 524288);

  fps_kernel    <<<BATCH, 1024, 0, stream>>>(xyz, new_xyz);
  knn_kernel    <<<BATCH * (NPOINT / 256), 256, 0, stream>>>(xyz, new_xyz, knn);
  gemm_wmma_kernel<<<MTOT / 128, 256, 0, stream>>>(px, xyz, new_xyz, knn, W, bias, h);
  bn_stage1     <<<512, 128, 0, stream>>>(h, psum, psq);
  bn_finalize   <<<1,   128, 0, stream>>>(psum, psq, gamma, beta, ab);
  maxpool_kernel<<<OUT_XYZ_OFF / 256, 256, 0, stream>>>(h, ab, out);
}